// FFTCausalConv_37769942401651
// MI455X (gfx1250) — compile-verified
//
#include <hip/hip_runtime.h>
#include <hip/hip_bf16.h>

#if __has_builtin(__builtin_amdgcn_global_load_async_to_lds_b128) && \
    __has_builtin(__builtin_amdgcn_s_wait_asynccnt)
#define HAS_ASYNC 1
#else
#define HAS_ASYNC 0
#endif

#define BB 4
#define LL 4096
#define DD 1024
#define D4 (DD / 4)      // 256 float4 lanes across channels
#define CC 64            // chunks along L
#define TT (LL / CC)     // 64 timesteps per chunk
#define ROWS 8           // rows per LDS tile (8 * 4KB = 32KB per buffer)
#define PFD 8            // prefetch distance for fallback path

typedef __attribute__((__vector_size__(16))) int vint4;

#if HAS_ASYNC
// Builtin signature (from clang diagnostic): pointee type is v4i (int vector
// of 16 bytes); global side is AS1, LDS side AS3. C-style casts perform the
// addrspacecast + pointee reinterpret in one step.
#define ALOAD_B128(gp, lp)                                                   \
    __builtin_amdgcn_global_load_async_to_lds_b128(                          \
        (__attribute__((address_space(1))) vint4*)(gp),                      \
        (__attribute__((address_space(3))) vint4*)(lp), 0, 0)
#define AWAIT() __builtin_amdgcn_s_wait_asynccnt(0)
#endif

__device__ __forceinline__ float4 fma4(float4 a, float4 s, float4 m, float4 x) {
    float4 r;
    r.x = fmaf(a.x, s.x, m.x * x.x);
    r.y = fmaf(a.y, s.y, m.y * x.y);
    r.z = fmaf(a.z, s.z, m.z * x.z);
    r.w = fmaf(a.w, s.w, m.w * x.w);
    return r;
}

// ---------------------------------------------------------------------------
// Probe kernel FIRST in the file so the disasm snippet shows how the async
// path lowered. Never launched.
// ---------------------------------------------------------------------------
extern "C" __global__ void aa_async_probe(const float* __restrict__ g,
                                          float* __restrict__ o) {
#if HAS_ASYNC
    __shared__ float4 buf[64];
    ALOAD_B128(((const float4*)g) + threadIdx.x, &buf[threadIdx.x]);
    AWAIT();
    float4 v = buf[threadIdx.x];
    ((float4*)o)[threadIdx.x] = v;
#else
    o[threadIdx.x] = g[threadIdx.x];
#endif
}

// K0: per-channel params. alpha = softplus(log_alpha), a = exp(-alpha),
// inv_norm = 1/max(||h||,1e-6), ||h||^2 = (1-a^{2L})/(1-a^2), aT = a^T.
__global__ void k0_params(const float* __restrict__ log_alpha,
                          float* __restrict__ pa,
                          float* __restrict__ pinv,
                          float* __restrict__ paT) {
    int d = blockIdx.x * blockDim.x + threadIdx.x;
    if (d >= DD) return;
    float la    = log_alpha[d];
    float alpha = log1pf(expf(la));
    float a     = expf(-alpha);
    float a2L   = expf(-2.0f * alpha * (float)LL);
    float nrm2  = (1.0f - a2L) / (1.0f - a * a);
    float nrm   = fmaxf(sqrtf(nrm2), 1e-6f);
    pa[d]   = a;
    pinv[d] = 1.0f / nrm;
    paT[d]  = expf(-alpha * (float)TT);
}

// K1: per-chunk local scan from zero state; emit chunk-end partial E[b][c][d4].
__global__ void k1_partials(const float4* __restrict__ x4,
                            const float4* __restrict__ pa4,
                            const float4* __restrict__ pinv4,
                            float4* __restrict__ E4) {
    const int d4 = threadIdx.x;          // 0..255
    const int c  = blockIdx.x;           // 0..CC-2
    const int b  = blockIdx.y;
    const float4 a   = pa4[d4];
    const float4 inv = pinv4[d4];
    float4 p = make_float4(0.f, 0.f, 0.f, 0.f);
    const float4* xp = x4 + (size_t)(b * LL + c * TT) * D4 + d4;

#if HAS_ASYNC
    __shared__ float4 tile[2][ROWS * D4];
    for (int r = 0; r < ROWS; ++r)
        ALOAD_B128(xp + (size_t)r * D4, &tile[0][r * D4 + d4]);
    AWAIT();
    const int NB = TT / ROWS;
    for (int t = 0; t < NB; ++t) {
        const int cur = t & 1, nxt = cur ^ 1;
        if (t + 1 < NB) {
            const float4* xn = xp + (size_t)(t + 1) * ROWS * D4;
            for (int r = 0; r < ROWS; ++r)
                ALOAD_B128(xn + (size_t)r * D4, &tile[nxt][r * D4 + d4]);
        }
#pragma unroll
        for (int r = 0; r < ROWS; ++r) {
            float4 xv = tile[cur][r * D4 + d4];   // ds_load_b128, own slot only
            p = fma4(a, p, inv, xv);
        }
        AWAIT();                                   // next buffer resident
    }
#else
#pragma unroll 8
    for (int i = 0; i < TT; ++i) {
        __builtin_prefetch(xp + (size_t)(i + PFD) * D4, 0, 1);
        float4 xv = xp[(size_t)i * D4];
        p = fma4(a, p, inv, xv);
    }
#endif
    E4[(size_t)(b * CC + c) * D4 + d4] = p;
}

// K2: tiny cross-chunk scan per (b,d4): S[0]=carry; S[c]=E[c-1]+a^T*S[c-1].
__global__ void k2_chunkscan(const float4* __restrict__ E4,
                             const float4* __restrict__ carry4,
                             const float4* __restrict__ paT4,
                             float4* __restrict__ S4) {
    int t = blockIdx.x * blockDim.x + threadIdx.x;
    if (t >= BB * D4) return;
    const int b  = t / D4;
    const int d4 = t % D4;
    const float4 aT = paT4[d4];
    float4 s = carry4[(size_t)b * D4 + d4];
    for (int c = 0; c < CC; ++c) {
        S4[(size_t)(b * CC + c) * D4 + d4] = s;
        if (c < CC - 1) {
            float4 e = E4[(size_t)(b * CC + c) * D4 + d4];
            s.x = fmaf(aT.x, s.x, e.x);
            s.y = fmaf(aT.y, s.y, e.y);
            s.z = fmaf(aT.z, s.z, e.z);
            s.w = fmaf(aT.w, s.w, e.w);
        }
    }
}

// K3: final pass — re-scan chunk with correct incoming state, write y,
// last chunk also writes new_carry (= y[:, -1, :]).
__global__ void k3_final(const float4* __restrict__ x4,
                         const float4* __restrict__ S4,
                         const float4* __restrict__ pa4,
                         const float4* __restrict__ pinv4,
                         float4* __restrict__ y4,
                         float4* __restrict__ carry_out4) {
    const int d4 = threadIdx.x;
    const int c  = blockIdx.x;           // 0..CC-1
    const int b  = blockIdx.y;
    const float4 a   = pa4[d4];
    const float4 inv = pinv4[d4];
    float4 s = S4[(size_t)(b * CC + c) * D4 + d4];
    const float4* xp = x4 + (size_t)(b * LL + c * TT) * D4 + d4;
    float4*       yp = y4 + (size_t)(b * LL + c * TT) * D4 + d4;

#if HAS_ASYNC
    __shared__ float4 tile[2][ROWS * D4];
    for (int r = 0; r < ROWS; ++r)
        ALOAD_B128(xp + (size_t)r * D4, &tile[0][r * D4 + d4]);
    AWAIT();
    const int NB = TT / ROWS;
    for (int t = 0; t < NB; ++t) {
        const int cur = t & 1, nxt = cur ^ 1;
        if (t + 1 < NB) {
            const float4* xn = xp + (size_t)(t + 1) * ROWS * D4;
            for (int r = 0; r < ROWS; ++r)
                ALOAD_B128(xn + (size_t)r * D4, &tile[nxt][r * D4 + d4]);
        }
        float4* yb = yp + (size_t)t * ROWS * D4;
#pragma unroll
        for (int r = 0; r < ROWS; ++r) {
            float4 xv = tile[cur][r * D4 + d4];
            s = fma4(a, s, inv, xv);
            yb[(size_t)r * D4] = s;               // coalesced 512B/wave store
        }
        AWAIT();
    }
#else
#pragma unroll 8
    for (int i = 0; i < TT; ++i) {
        __builtin_prefetch(xp + (size_t)(i + PFD) * D4, 0, 1);
        float4 xv = xp[(size_t)i * D4];
        s = fma4(a, s, inv, xv);
        yp[(size_t)i * D4] = s;
    }
#endif
    if (c == CC - 1) {
        carry_out4[(size_t)b * D4 + d4] = s;
    }
}

extern "C" void kernel_launch(void* const* d_in, const int* in_sizes, int n_in,
                              void* d_out, int out_size, void* d_ws, size_t ws_size,
                              hipStream_t stream) {
    const float* x     = (const float*)d_in[0];   // (B, L, D) f32
    const float* carry = (const float*)d_in[1];   // (B, D)    f32
    const float* la    = (const float*)d_in[2];   // (D,)      f32
    float* out = (float*)d_out;                   // y (B,L,D) then new_carry (B,D)

    // Workspace (floats): pa[DD] | pinv[DD] | paT[DD] | pad->4096 | E | S  (~2.07MB)
    float* wsf  = (float*)d_ws;
    float* pa   = wsf;
    float* pinv = wsf + DD;
    float* paT  = wsf + 2 * DD;
    float* E    = wsf + 4096;
    float* S    = E + (size_t)BB * CC * DD;

    k0_params<<<(DD + 255) / 256, 256, 0, stream>>>(la, pa, pinv, paT);

    k1_partials<<<dim3(CC - 1, BB), 256, 0, stream>>>(
        (const float4*)x, (const float4*)pa, (const float4*)pinv, (float4*)E);

    k2_chunkscan<<<(BB * D4 + 255) / 256, 256, 0, stream>>>(
        (const float4*)E, (const float4*)carry, (const float4*)paT, (float4*)S);

    k3_final<<<dim3(CC, BB), 256, 0, stream>>>(
        (const float4*)x, (const float4*)S, (const float4*)pa, (const float4*)pinv,
        (float4*)out, (float4*)(out + (size_t)BB * LL * DD));
}